// ALIFNeuron_32719060861204
// MI455X (gfx1250) — compile-verified
//
#include <hip/hip_runtime.h>
#include <hip/hip_bf16.h>

// ALIF e-prop forward: B=32, T=48, I=128, H=256, O=64
#define Bz 32
#define Tz 48
#define Iz 128
#define Hz 256
#define Oz 64

#define ALPHA 0.9512294245007140f   // exp(-1/20)
#define RHO   0.9995001249791693f   // exp(-1/2000)
#define KAPPA 0.9512294245007140f   // exp(-1/20)
#define BETA  0.07f
#define V_TH  0.6f
#define GAMMA 0.3f

typedef __attribute__((ext_vector_type(2))) float v2f;
typedef __attribute__((ext_vector_type(8))) float v8f;

#if __has_builtin(__builtin_amdgcn_wmma_f32_16x16x4_f32)
#define HAVE_WMMA_F32 1
#endif

__device__ __forceinline__ v8f wmma_f32(v2f a, v2f b, v8f c) {
#ifdef HAVE_WMMA_F32
  // (neg_a, A, neg_b, B, c_mod, C, reuse_a, reuse_b) -> v_wmma_f32_16x16x4_f32
  return __builtin_amdgcn_wmma_f32_16x16x4_f32(false, a, false, b, (short)0, c,
                                               false, false);
#else
  return c;  // host pass / builtin missing: placeholder (histogram shows wmma=0)
#endif
}

// ---------------------------------------------------------------------------
// Phase 1: the sequential T-scan (one workgroup = 32 wave32 on one WGP).
//   Per step: i_t = x_t@w_in^T + z@w_rec^T via fp32 WMMA (32 output tiles,
//   one per wave), pointwise ALIF update (v,a,tr_rec in registers; z in LDS),
//   readout = KAPPA*readout + z@w_out^T via WMMA (8 tiles, waves 0..7).
//   Emits psi[t] and tr_rec[t] tables for phase 2.
// ---------------------------------------------------------------------------
__global__ __launch_bounds__(1024)
void alif_phase1(const float* __restrict__ x,      // (B,T,I)
                 const float* __restrict__ w_in,   // (H,I)
                 const float* __restrict__ w_rec,  // (H,H)
                 const float* __restrict__ w_out,  // (O,H)
                 float* __restrict__ ro_out,       // (B,T,O)
                 float* __restrict__ ws_psi,       // (T, B*H)
                 float* __restrict__ ws_tr)        // (T, B*H)
{
  __shared__ float zL[Bz * Hz];    // spikes z (current step)        32 KB
  __shared__ float itmp[Bz * Hz];  // i_t staging                    32 KB

  const int tid    = threadIdx.x;
  const int wv     = tid >> 5;
  const int lane   = tid & 31;
  const int lane15 = lane & 15;
  const int khalf  = (lane >> 4) << 1;   // 0 or 2: K sub-column per lane half
  const int rsel   = (lane >> 4) << 3;   // 0 or 8: C/D row offset per lane half

  // init z = 0
#pragma unroll
  for (int r = 0; r < 8; ++r) zL[tid + r * 1024] = 0.f;

  // per-thread register state for the pointwise stage (fixed element mapping)
  float vS[8], aS[8], trS[8];
#pragma unroll
  for (int r = 0; r < 8; ++r) { vS[r] = 0.f; aS[r] = 0.f; trS[r] = 0.f; }

  // stage-A tile: 2 M-tiles (batch) x 16 N-tiles (hidden) = 32 tiles
  const int aMt = wv >> 4;
  const int aNt = wv & 15;
  const int a_b = aMt * 16 + lane15;  // batch row for A fragment
  const int a_n = aNt * 16 + lane15;  // hidden col for B fragment

  // stage-C tile: 2 M-tiles (batch) x 4 N-tiles (output) = 8 tiles (waves 0..7)
  const int cMt = (wv >> 2) & 1;
  const int cNt = wv & 3;
  const int c_b = cMt * 16 + lane15;
  const int c_o = cNt * 16 + lane15;

  v8f ro = {0.f, 0.f, 0.f, 0.f, 0.f, 0.f, 0.f, 0.f};

  __syncthreads();

  for (int t = 0; t < Tz; ++t) {
    // ---- stage A: i_t tiles via WMMA f32 16x16x4 -------------------------
    {
      v8f c = {0.f, 0.f, 0.f, 0.f, 0.f, 0.f, 0.f, 0.f};
      const float* xrow = x + ((size_t)a_b * Tz + t) * Iz;
      const float* wrow = w_in + (size_t)a_n * Iz;
#pragma unroll 4
      for (int k0 = 0; k0 < Iz; k0 += 4) {
        const int kb = k0 + khalf;
        float2 ax = *reinterpret_cast<const float2*>(xrow + kb);
        float2 bx = *reinterpret_cast<const float2*>(wrow + kb);
        v2f av, bv;
        av[0] = ax.x; av[1] = ax.y;
        bv[0] = bx.x; bv[1] = bx.y;
        c = wmma_f32(av, bv, c);
      }
      const float* zrow = zL + a_b * Hz;
      const float* rrow = w_rec + (size_t)a_n * Hz;
#pragma unroll 4
      for (int k0 = 0; k0 < Hz; k0 += 4) {
        const int kb = k0 + khalf;
        float2 az = *reinterpret_cast<const float2*>(zrow + kb);
        float2 bx = *reinterpret_cast<const float2*>(rrow + kb);
        v2f av, bv;
        av[0] = az.x; av[1] = az.y;
        bv[0] = bx.x; bv[1] = bx.y;
        c = wmma_f32(av, bv, c);
      }
#pragma unroll
      for (int j = 0; j < 8; ++j) {
        const int brow = aMt * 16 + j + rsel;
        itmp[brow * Hz + a_n] = c[j];
      }
    }
    __syncthreads();

    // ---- stage B: pointwise ALIF neuron update (8192 elems / 1024 thr) ---
    {
#pragma unroll
      for (int r = 0; r < 8; ++r) {
        const int e = tid + r * 1024;
        const float zp  = zL[e];
        const float Ath = V_TH + BETA * aS[r];
        float vt        = ALPHA * vS[r] + itmp[e] - zp * V_TH;
        const float dist = vt - Ath;
        const float psi  = (GAMMA / V_TH) * fmaxf(0.f, 1.f - fabsf(dist) * (1.f / V_TH));
        const float zn   = (dist > 0.f) ? 1.f : 0.f;
        vt    = (zn > 0.f) ? dist : vt;      // v*(1-z) + (v-A)*z
        aS[r]  = RHO * aS[r] + zn;
        trS[r] = ALPHA * trS[r] + zp;        // tr_rec uses z_prev
        vS[r]  = vt;
        zL[e]  = zn;
        ws_psi[(size_t)t * (Bz * Hz) + e] = psi;
        ws_tr [(size_t)t * (Bz * Hz) + e] = trS[r];
      }
    }
    __syncthreads();

    // ---- stage C: readout = KAPPA*readout + z_next @ w_out^T (waves 0..7)
    if (wv < 8) {
#pragma unroll
      for (int j = 0; j < 8; ++j) ro[j] *= KAPPA;
      const float* zrow = zL + c_b * Hz;
      const float* orow = w_out + (size_t)c_o * Hz;
#pragma unroll 4
      for (int k0 = 0; k0 < Hz; k0 += 4) {
        const int kb = k0 + khalf;
        float2 az = *reinterpret_cast<const float2*>(zrow + kb);
        float2 bx = *reinterpret_cast<const float2*>(orow + kb);
        v2f av, bv;
        av[0] = az.x; av[1] = az.y;
        bv[0] = bx.x; bv[1] = bx.y;
        ro = wmma_f32(av, bv, ro);
      }
#pragma unroll
      for (int j = 0; j < 8; ++j) {
        const int brow = cMt * 16 + j + rsel;
        ro_out[((size_t)brow * Tz + t) * Oz + c_o] = ro[j];
      }
    }
  }
}

// ---------------------------------------------------------------------------
// Phase 2: eligibility traces — fully parallel per-synapse scans.
// State (tr, eps, fe) lives in registers; psi broadcast via LDS; source
// tables are tiny and L2-resident.
// ---------------------------------------------------------------------------
__global__ __launch_bounds__(Iz)
void alif_phase2_in(const float* __restrict__ x,       // (B,T,I)
                    const float* __restrict__ ws_psi,  // (T, B*H)
                    float* __restrict__ fe_in)         // (B,H,I)
{
  __shared__ float psiS[Tz];
  const int bh = blockIdx.x;       // b*H + h
  const int b  = bh >> 8;          // H = 256
  const int i  = threadIdx.x;
  if (i < Tz) psiS[i] = ws_psi[(size_t)i * (Bz * Hz) + bh];
  __syncthreads();

  float tr = 0.f, eps = 0.f, fe = 0.f;
  const float* xp = x + (size_t)b * Tz * Iz + i;
#pragma unroll 4
  for (int t = 0; t < Tz; ++t) {
    tr = ALPHA * tr + xp[t * Iz];               // tr_in scan (recomputed)
    const float p = psiS[t];
    eps = (RHO - p * BETA) * eps + p * tr;
    fe  = KAPPA * fe + p * (tr - BETA * eps);
  }
  fe_in[(size_t)bh * Iz + i] = fe;
}

__global__ __launch_bounds__(Hz)
void alif_phase2_rec(const float* __restrict__ ws_tr,   // (T, B*H)
                     const float* __restrict__ ws_psi,  // (T, B*H)
                     float* __restrict__ fe_rec)        // (B,H,H)
{
  __shared__ float psiS[Tz];
  const int bh = blockIdx.x;       // b*H + h (postsynaptic)
  const int b  = bh >> 8;
  const int j  = threadIdx.x;      // presynaptic index
  if (j < Tz) psiS[j] = ws_psi[(size_t)j * (Bz * Hz) + bh];
  __syncthreads();

  float eps = 0.f, fe = 0.f;
  const float* tp = ws_tr + (size_t)b * Hz + j;
#pragma unroll 4
  for (int t = 0; t < Tz; ++t) {
    const float tr = tp[(size_t)t * (Bz * Hz)];
    const float p  = psiS[t];
    eps = (RHO - p * BETA) * eps + p * tr;
    fe  = KAPPA * fe + p * (tr - BETA * eps);
  }
  fe_rec[(size_t)bh * Hz + j] = fe;
}

extern "C" void kernel_launch(void* const* d_in, const int* in_sizes, int n_in,
                              void* d_out, int out_size, void* d_ws, size_t ws_size,
                              hipStream_t stream) {
  const float* x     = (const float*)d_in[0];   // (B,T,I)
  const float* w_in  = (const float*)d_in[1];   // (H,I)
  const float* w_rec = (const float*)d_in[2];   // (H,H)
  const float* w_out = (const float*)d_in[3];   // (O,H)

  float* out    = (float*)d_out;
  float* ro_out = out;                                 // readout_seq (B,T,O)
  float* fe_in  = out + (size_t)Bz * Tz * Oz;          // (B,H,I)
  float* fe_rec = fe_in + (size_t)Bz * Hz * Iz;        // (B,H,H)

  float* ws     = (float*)d_ws;
  float* ws_psi = ws;                                  // T*B*H floats (1.5 MB)
  float* ws_tr  = ws + (size_t)Tz * Bz * Hz;           // T*B*H floats (1.5 MB)

  alif_phase1<<<1, 1024, 0, stream>>>(x, w_in, w_rec, w_out, ro_out, ws_psi, ws_tr);
  alif_phase2_in <<<Bz * Hz, Iz, 0, stream>>>(x, ws_psi, fe_in);
  alif_phase2_rec<<<Bz * Hz, Hz, 0, stream>>>(ws_tr, ws_psi, fe_rec);
}